// CLIP_Inplanted_groupPNmixAfterConv_groupMaxNensembleOut_73418170957876
// MI455X (gfx1250) — compile-verified
//
#include <hip/hip_runtime.h>
#include <stdint.h>

typedef float v2f __attribute__((ext_vector_type(2)));
typedef float v8f __attribute__((ext_vector_type(8)));
typedef unsigned int v4u __attribute__((ext_vector_type(4)));
typedef int v8i __attribute__((ext_vector_type(8)));
typedef int v4i __attribute__((ext_vector_type(4)));

#define HW 289
#define CCH 1024
#define BB 64
#define SPB 18496u              /* B*H*W */
#define STATN (18496u * 255u)   /* sum over levels of B*g*H*W */
#define EPSF 1e-5f
#define CHUNK 4624              /* 16 channels * 289 */

__device__ __forceinline__ unsigned lvloff(int i) {
  return 18496u * ((1u << i) - 1u);
}

/* ---------------- TDM helper: 1-D contiguous global -> LDS copy ------------- */
__device__ __forceinline__ void tdm_load_1d(float* lds_dst, const float* gsrc,
                                            unsigned nelem) {
  unsigned ldsOff = (unsigned)(uintptr_t)(void*)lds_dst;  /* low 32 bits = LDS byte offset */
  unsigned long long ga = (unsigned long long)(uintptr_t)gsrc;
  /* D# group0: count=1 | lds_addr | global_addr[56:0] | type=2 */
  v4u g0;
  g0[0] = 1u;
  g0[1] = ldsOff;
  g0[2] = (unsigned)(ga & 0xffffffffull);
  g0[3] = ((unsigned)((ga >> 32) & 0x01ffffffull)) | (2u << 30);
  /* D# group1: data_size=4B (code 2); tensor_dim0=nelem; tensor_dim1=1;
     tile_dim0=nelem (1-D tile); stride0=nelem */
  v8i g1;
  g1[0] = (int)(2u << 16);
  g1[1] = (int)((nelem & 0xffffu) << 16);
  g1[2] = (int)(((nelem >> 16) & 0xffffu) | (1u << 16));
  g1[3] = (int)((nelem & 0xffffu) << 16);
  g1[4] = 0;
  g1[5] = (int)nelem;
  g1[6] = 0;
  g1[7] = 0;
  v4i g2 = {0, 0, 0, 0};
  v4i g3 = {0, 0, 0, 0};
  v8i g4 = {0, 0, 0, 0, 0, 0, 0, 0};
  __builtin_amdgcn_tensor_load_to_lds(g0, g1, g2, g3, g4, 0);
}

/* ---------------- Pass 1: hierarchical group stats via f32 WMMA ------------- */
/* One wave per (b, 16-column spatial tile). Fine (8-channel) sums/sumsq are
   produced as 16x16 WMMA tiles (A = one-hot selector, exact), then a shfl
   gather feeds the 8-level streaming hierarchy to emit all 255 mean/std. */
__global__ __launch_bounds__(32) void stats_kernel(const float* __restrict__ x,
                                                   float* __restrict__ meanArr,
                                                   float* __restrict__ stdArr) {
  const int tile = blockIdx.x;   /* 0..18 */
  const int b = blockIdx.y;      /* 0..63 */
  const int lane = threadIdx.x;
  const int n = lane & 15;
  const int hi = lane >> 4;
  const int s0 = tile * 16;
  const int s = s0 + n;
  const int sc = (s < HW) ? s : (HW - 1); /* clamped load col (EXEC must stay full for WMMA) */
  const float* xb = x + (size_t)b * (CCH * HW);

  float accS[8], accQ[8];
#pragma unroll
  for (int i = 0; i < 8; ++i) { accS[i] = 0.f; accQ[i] = 0.f; }

  for (int t = 0; t < 8; ++t) {            /* 8 row-tiles of 16 fine groups */
    v8f Cs = {0.f, 0.f, 0.f, 0.f, 0.f, 0.f, 0.f, 0.f};
    v8f Cq = {0.f, 0.f, 0.f, 0.f, 0.f, 0.f, 0.f, 0.f};
    const int cbase = t * 128;
#pragma unroll 4
    for (int kk = 0; kk < 32; ++kk) {      /* K=4 steps over 128 channels */
      const int ch = cbase + kk * 4 + hi * 2;
      const float b0 = xb[(size_t)ch * HW + sc];
      const float b1 = xb[(size_t)(ch + 1) * HW + sc];
      const float av = (n == (kk >> 1)) ? 1.0f : 0.0f;
      v2f A = {av, av};
      v2f Bv = {b0, b1};
      v2f Bq = {b0 * b0, b1 * b1};
      Cs = __builtin_amdgcn_wmma_f32_16x16x4_f32(false, A, false, Bv, (short)0,
                                                 Cs, false, false);
      Cq = __builtin_amdgcn_wmma_f32_16x16x4_f32(false, A, false, Bq, (short)0,
                                                 Cq, false, false);
    }
    /* gather the 16 fine values of this column into lanes 0..15 */
    float colS[16], colQ[16];
#pragma unroll
    for (int p = 0; p < 8; ++p) {
      colS[p] = Cs[p];
      colQ[p] = Cq[p];
      colS[8 + p] = __shfl_down(Cs[p], 16, 32);
      colQ[8 + p] = __shfl_down(Cq[p], 16, 32);
    }
    const bool writer = (hi == 0) && (s < HW);
#pragma unroll
    for (int jl = 0; jl < 16; ++jl) {
      const int j = t * 16 + jl;           /* fine group 0..127 */
      const float s8 = colS[jl];
      const float q8 = colQ[jl];
#pragma unroll
      for (int i = 0; i < 8; ++i) {
        accS[i] += s8;
        accQ[i] += q8;
        const int period = 1 << (7 - i);
        if (((j + 1) & (period - 1)) == 0) {   /* group boundary: finalize */
          const float gc = (float)(1 << (10 - i));
          const float mn = accS[i] / gc;
          const float var = (accQ[i] - accS[i] * accS[i] / gc) / (gc - 1.0f);
          const float sd = sqrtf(fmaxf(var, 0.0f) + EPSF);
          if (writer) {
            const int k = j >> (7 - i);
            const unsigned idx =
                lvloff(i) + (unsigned)((((b << i) + k) * HW) + s);
            meanArr[idx] = mn;
            stdArr[idx] = sd;
          }
          accS[i] = 0.f;
          accQ[i] = 0.f;
        }
      }
    }
  }
}

/* ---------------- Pass 2: bake per-group mix coefficients ------------------- */
/* mix = 0.5*((x-m1)/s1*s2 + m2 + (xp-m2)/s2*s1 + m1) = A*x + B*xp + C */
__global__ __launch_bounds__(256) void coef_kernel(
    const float* __restrict__ meanArr, const float* __restrict__ stdArr,
    const int* __restrict__ perm, float* __restrict__ cfA,
    float* __restrict__ cfB, float* __restrict__ cfC) {
  const unsigned idx = blockIdx.x * 256u + threadIdx.x;
  if (idx >= STATN) return;
  int lvl = 0;
  unsigned e = idx;
#pragma unroll
  for (int i = 7; i >= 0; --i) {
    if (idx >= lvloff(i)) { lvl = i; e = idx - lvloff(i); break; }
  }
  const unsigned gsz = (1u << lvl) * (unsigned)HW; /* elems per batch at this level */
  const unsigned b = e / gsz;
  const unsigned r = e - b * gsz;
  const unsigned pb = (unsigned)perm[b];
  const unsigned pidx = lvloff(lvl) + pb * gsz + r;
  const float m1 = meanArr[idx], s1 = stdArr[idx];
  const float m2 = meanArr[pidx], s2 = stdArr[pidx];
  const float r12 = s2 / s1;
  const float r21 = s1 / s2;
  cfA[idx] = 0.5f * r12;
  cfB[idx] = 0.5f * r21;
  cfC[idx] = 0.5f * (m1 + m2 - m1 * r12 - m2 * r21);
}

/* ---------------- Pass 3: per-level variance scores (TDM-staged) ------------ */
__global__ __launch_bounds__(256) void score_kernel(
    const float* __restrict__ x, const int* __restrict__ perm,
    const float* __restrict__ cfA, const float* __restrict__ cfB,
    const float* __restrict__ cfC, double* __restrict__ acc) {
  __shared__ float smem[2 * CHUNK];
  __shared__ double lacc[16];
  const int b = blockIdx.x >> 6;
  const int c0 = (blockIdx.x & 63) << 4;
  const int tid = threadIdx.x;
  if (tid < 16) lacc[tid] = 0.0;
  const int pb = perm[b];
  const float* gx = x + ((size_t)b * CCH + c0) * HW;
  const float* gxp = x + ((size_t)pb * CCH + c0) * HW;
  if (tid < 32) { /* wave 0 drives the Tensor Data Mover */
    tdm_load_1d(&smem[0], gx, CHUNK);
    tdm_load_1d(&smem[CHUNK], gxp, CHUNK);
    __builtin_amdgcn_s_wait_tensorcnt((short)0);
  }
  __syncthreads();

  float ls[8], lq[8];
#pragma unroll
  for (int i = 0; i < 8; ++i) { ls[i] = 0.f; lq[i] = 0.f; }

  for (int t = tid; t < CHUNK; t += 256) {
    const unsigned cl = (unsigned)t / HW;
    const unsigned s = (unsigned)t - cl * HW;
    const unsigned c = (unsigned)c0 + cl;
    const float xv = smem[t];
    const float xpv = smem[CHUNK + t];
#pragma unroll
    for (int i = 0; i < 8; ++i) {
      const unsigned g = 1u << i;
      const unsigned k = c >> (10 - i);
      const unsigned cix =
          lvloff(i) + (((unsigned)b * g + k) * (unsigned)HW) + s;
      const float m = fmaf(cfA[cix], xv, fmaf(cfB[cix], xpv, cfC[cix]));
      ls[i] += m;
      lq[i] = fmaf(m, m, lq[i]);
    }
  }
#pragma unroll
  for (int i = 0; i < 8; ++i) {
    float a = ls[i], q = lq[i];
#pragma unroll
    for (int o = 16; o > 0; o >>= 1) {
      a += __shfl_down(a, o, 32);
      q += __shfl_down(q, o, 32);
    }
    if ((tid & 31) == 0) {
      atomicAdd(&lacc[i], (double)a);
      atomicAdd(&lacc[8 + i], (double)q);
    }
  }
  __syncthreads();
  if (tid < 16) atomicAdd(&acc[tid], lacc[tid]);
}

/* ---------------- Pass 4: top-3 selection ---------------------------------- */
__global__ void topk_kernel(const double* __restrict__ acc,
                            int* __restrict__ sel) {
  if (threadIdx.x == 0) {
    const double N = 18939904.0; /* B*C*H*W */
    double var[8];
    bool used[8];
#pragma unroll
    for (int i = 0; i < 8; ++i) {
      const double s = acc[i], q = acc[8 + i];
      var[i] = (q - s * s / N) / (N - 1.0);
      used[i] = false;
    }
    for (int r = 0; r < 3; ++r) {
      int bi = 0;
      double bv = -1.0e300;
      for (int i = 0; i < 8; ++i) {
        if (!used[i] && var[i] > bv) { bv = var[i]; bi = i; } /* ties -> lowest idx */
      }
      used[bi] = true;
      sel[r] = bi;
    }
  }
}

/* ---------------- Pass 5: average of the 3 winning mixes (TDM-staged) ------- */
__global__ __launch_bounds__(256) void out_kernel(
    const float* __restrict__ x, const int* __restrict__ perm,
    const float* __restrict__ cfA, const float* __restrict__ cfB,
    const float* __restrict__ cfC, const int* __restrict__ sel,
    float* __restrict__ out) {
  __shared__ float smem[2 * CHUNK];
  const int b = blockIdx.x >> 6;
  const int c0 = (blockIdx.x & 63) << 4;
  const int tid = threadIdx.x;
  const int pb = perm[b];
  const float* gx = x + ((size_t)b * CCH + c0) * HW;
  const float* gxp = x + ((size_t)pb * CCH + c0) * HW;
  if (tid < 32) {
    tdm_load_1d(&smem[0], gx, CHUNK);
    tdm_load_1d(&smem[CHUNK], gxp, CHUNK);
    __builtin_amdgcn_s_wait_tensorcnt((short)0);
  }
  __syncthreads();

  const int i0 = sel[0], i1 = sel[1], i2 = sel[2];
  float* gout = out + ((size_t)b * CCH + c0) * HW;

  for (int t = tid; t < CHUNK; t += 256) {
    const unsigned cl = (unsigned)t / HW;
    const unsigned s = (unsigned)t - cl * HW;
    const unsigned c = (unsigned)c0 + cl;
    const float xv = smem[t];
    const float xpv = smem[CHUNK + t];
    float av = 0.f;
    int levels[3] = {i0, i1, i2};
#pragma unroll
    for (int r = 0; r < 3; ++r) {
      const int i = levels[r];
      const unsigned g = 1u << i;
      const unsigned k = c >> (10 - i);
      const unsigned cix =
          lvloff(i) + (((unsigned)b * g + k) * (unsigned)HW) + s;
      av += fmaf(cfA[cix], xv, fmaf(cfB[cix], xpv, cfC[cix]));
    }
    gout[t] = av * (1.0f / 3.0f);
  }
}

__global__ void zero_acc(double* acc) {
  if (threadIdx.x < 16) acc[threadIdx.x] = 0.0;
}

extern "C" void kernel_launch(void* const* d_in, const int* in_sizes, int n_in,
                              void* d_out, int out_size, void* d_ws,
                              size_t ws_size, hipStream_t stream) {
  (void)in_sizes; (void)n_in; (void)out_size; (void)ws_size;
  const float* x = (const float*)d_in[0];
  const int* perm = (const int*)d_in[1];
  float* out = (float*)d_out;

  float* ws = (float*)d_ws;
  float* meanA = ws;                 /* STATN floats */
  float* stdA = meanA + STATN;       /* STATN floats */
  float* cfA = stdA + STATN;         /* STATN floats */
  float* cfB = cfA + STATN;          /* STATN floats */
  float* cfC = cfB + STATN;          /* STATN floats */
  double* acc = (double*)(cfC + STATN); /* 16 doubles (8-byte aligned) */
  int* sel = (int*)(acc + 16);          /* 3 ints */

  zero_acc<<<1, 32, 0, stream>>>(acc);
  stats_kernel<<<dim3(19, 64), 32, 0, stream>>>(x, meanA, stdA);
  coef_kernel<<<(STATN + 255u) / 256u, 256, 0, stream>>>(meanA, stdA, perm,
                                                         cfA, cfB, cfC);
  score_kernel<<<4096, 256, 0, stream>>>(x, perm, cfA, cfB, cfC, acc);
  topk_kernel<<<1, 32, 0, stream>>>(acc, sel);
  out_kernel<<<4096, 256, 0, stream>>>(x, perm, cfA, cfB, cfC, sel, out);
}